// SlaterPooling_22608707846150
// MI455X (gfx1250) — compile-verified
//
#include <hip/hip_runtime.h>
#include <stdint.h>

// Slater determinant pooling for MI455X (gfx1250).
// out[b, c] = det( X[b, 0:8, cup[c,:]] ) * det( X[b, 8:16, cdown[c,:]] )
//
// Roofline: 32 MB input read once -> ~1.6 us HBM floor; compute is 2.1M
// pivoted 8x8 LU factorizations (~0.7 GFLOP of sequential scalar work) ->
// VALU-bound. CDNA5 leverage: TDM (TENSOR_LOAD_TO_LDS + s_wait_tensorcnt)
// for the contiguous 8 KB X tile, ASYNCcnt lane-copies for the index
// tables, conflict-free LDS gathers, wave32 VALU with VOPD dual-issue.
// WMMA is deliberately not used: the op has no 16x16 matmul structure.

#define NUP    8
#define NMO    32
#define NCONF  64
#define NBATCH 16384
#define BPB    4     // batches per block
#define NTH    256   // 8 wave32 waves; BPB*NCONF threads

typedef unsigned int u32x4 __attribute__((ext_vector_type(4)));
typedef unsigned int u32x8 __attribute__((ext_vector_type(8)));

struct SMem {
  float X[BPB][16][NMO];   // 8192 B: 4 batches x 16 rows x 32 MOs
  int   cup[NCONF][NUP];   // 2048 B
  int   cdn[NCONF][NUP];   // 2048 B
};                         // 12 KB total (WGP has 320 KB)

// Async copy 16 bytes/lane from global to LDS (ASYNCcnt path).
// VDST carries the per-lane LDS byte offset (ISA 15.18: added to the LDS base);
// low 32 bits of a shared-aperture generic pointer == LDS offset.
__device__ __forceinline__ void async_b128_to_lds(const void* g, void* l) {
  unsigned loff = (unsigned)(uintptr_t)l;
  asm volatile("global_load_async_to_lds_b128 %0, %1, off"
               :: "v"(loff), "v"(g) : "memory");
}

__device__ __forceinline__ void wait_async0() {
#if __has_builtin(__builtin_amdgcn_s_wait_asynccnt)
  __builtin_amdgcn_s_wait_asynccnt(0);
#else
  asm volatile("s_wait_asynccnt 0x0" ::: "memory");
#endif
}

__device__ __forceinline__ void wait_tensor0() {
#if __has_builtin(__builtin_amdgcn_s_wait_tensorcnt)
  __builtin_amdgcn_s_wait_tensorcnt(0);
#else
  asm volatile("s_wait_tensorcnt 0x0" ::: "memory");
#endif
}

// One-shot TDM DMA: contiguous `elems` f32 from global -> LDS (TENSORcnt).
// D# per ISA cdna5 ch.8: group0 = {count=1, lds_addr, global_addr(57b),
// type=2}; group1 = {data_size=4B, tensor_dim0=tile_dim0=elems, dim1=1,
// stride0=elems}; <=2D tensor -> 2-SGPR-group form (VADDR2/3 = NULL).
__device__ __forceinline__ void tdm_load_1d(const void* g, void* l, unsigned elems) {
  uint64_t ga   = (uint64_t)(uintptr_t)g;
  unsigned loff = (unsigned)(uintptr_t)l;
  u32x4 g0;
  g0.x = 1u;                                             // count=1, no gather
  g0.y = loff;                                           // lds_addr (bytes)
  g0.z = (unsigned)ga;                                   // global_addr[31:0]
  g0.w = ((unsigned)(ga >> 32) & 0x01FFFFFFu)            // global_addr[56:32]
         | 0x80000000u;                                  // type=2 ("image")
  u32x8 g1;
  g1.s0 = 0x00020000u;          // data_size=2 (4B); mask/barrier/iter/pad=0
  g1.s1 = (elems & 0xFFFFu) << 16;                       // tensor_dim0[15:0]
  g1.s2 = ((elems >> 16) & 0xFFFFu) | (1u << 16);        // dim0[31:16], dim1=1
  g1.s3 = (elems & 0xFFFFu) << 16;                       // tile_dim0 (<=65535)
  g1.s4 = 1u;                                            // tile_dim1=1, dim2=0
  g1.s5 = elems;                                         // dim0_stride[31:0]
  g1.s6 = 0u;                                            // stride hi, dim1_stride
  g1.s7 = 0u;
  asm volatile("tensor_load_to_lds %0, %1" :: "s"(g0), "s"(g1) : "memory");
}

// Determinant of register-resident 8x8 via LU with partial pivoting
// (compare-exchange chain; only columns j>=k swapped; sign tracked).
// Zero/denormal pivot -> multiplier 0, so singular matrices (duplicated
// selected columns) give det ~ 0 instead of NaN.
__device__ __forceinline__ float det8(float M[8][8]) {
  float det = 1.0f;
#pragma unroll
  for (int k = 0; k < 8; ++k) {
#pragma unroll
    for (int r = k + 1; r < 8; ++r) {
      bool sw = __builtin_fabsf(M[r][k]) > __builtin_fabsf(M[k][k]);
      det = sw ? -det : det;
#pragma unroll
      for (int j = k; j < 8; ++j) {
        float a = M[k][j], b = M[r][j];
        M[k][j] = sw ? b : a;
        M[r][j] = sw ? a : b;
      }
    }
    float p = M[k][k];
    det *= p;
#if __has_builtin(__builtin_amdgcn_rcpf)
    float inv = (__builtin_fabsf(p) > 1e-30f) ? __builtin_amdgcn_rcpf(p) : 0.0f;
#else
    float inv = (__builtin_fabsf(p) > 1e-30f) ? (1.0f / p) : 0.0f;
#endif
#pragma unroll
    for (int r = k + 1; r < 8; ++r) {
      float f = M[r][k] * inv;
#pragma unroll
      for (int j = k + 1; j < 8; ++j)
        M[r][j] = __builtin_fmaf(-f, M[k][j], M[r][j]);
    }
  }
  return det;
}

__global__ __launch_bounds__(NTH) void slater_det_kernel(
    const float* __restrict__ inp,   // [NBATCH,16,NMO]
    const int*   __restrict__ cup,   // [NCONF,8]
    const int*   __restrict__ cdn,   // [NCONF,8]
    float*       __restrict__ out) { // [NBATCH,NCONF]
  __shared__ SMem sm;
  const unsigned t   = threadIdx.x;
  const unsigned blk = blockIdx.x;

  // ---- Stage X tile (8 KB) via TDM; index tables (2+2 KB) via ASYNCcnt ----
  const char* gX = (const char*)inp + (size_t)blk * (BPB * 16 * NMO * sizeof(float));
  if (t < 32) {  // one wave issues the DMA (TDM ignores EXEC -> must branch)
    tdm_load_1d(gX, &sm.X[0][0][0], BPB * 16 * NMO);
    wait_tensor0();
  }
  if (t < 128) {   // wave-uniform split (waves 0-3 vs 4-7)
    async_b128_to_lds((const char*)cup + t * 16,
                      (char*)&sm.cup[0][0] + t * 16);          // 2048 B
  } else {
    async_b128_to_lds((const char*)cdn + (t - 128) * 16,
                      (char*)&sm.cdn[0][0] + (t - 128) * 16);  // 2048 B
  }
  wait_async0();
  __syncthreads();

  // ---- One (batch, config) pair per thread ----
  const int c  = t & 63;   // lanes of a wave share bl => each LDS gather is
  const int bl = t >> 6;   // broadcast (equal idx) or distinct-bank (idx<32)

  float dprod = 1.0f;
#pragma unroll 1           // one det8 instantiation (I$ friendly)
  for (int s = 0; s < 2; ++s) {
    const int*   ci = (s == 0) ? &sm.cup[c][0] : &sm.cdn[c][0];
    const float* Xr = &sm.X[bl][s * NUP][0];
    int idx[8];
#pragma unroll
    for (int j = 0; j < 8; ++j) idx[j] = ci[j];
    float M[8][8];
#pragma unroll
    for (int i = 0; i < 8; ++i)
#pragma unroll
      for (int j = 0; j < 8; ++j)
        M[i][j] = Xr[i * NMO + idx[j]];
    dprod *= det8(M);
  }

  // b = blk*BPB + bl; out index b*64 + c == blk*256 + t  (coalesced)
  out[(size_t)blk * NTH + t] = dprod;
}

extern "C" void kernel_launch(void* const* d_in, const int* in_sizes, int n_in,
                              void* d_out, int out_size, void* d_ws, size_t ws_size,
                              hipStream_t stream) {
  (void)in_sizes; (void)n_in; (void)out_size; (void)d_ws; (void)ws_size;
  const float* inp = (const float*)d_in[0];
  const int*   cup = (const int*)d_in[1];
  const int*   cdn = (const int*)d_in[2];
  float*       out = (float*)d_out;
  slater_det_kernel<<<dim3(NBATCH / BPB), dim3(NTH), 0, stream>>>(inp, cup, cdn, out);
}